// incremental_NCM_classifier_43868795961907
// MI455X (gfx1250) — compile-verified
//
#include <hip/hip_runtime.h>

typedef float v2f __attribute__((ext_vector_type(2)));
typedef float v8f __attribute__((ext_vector_type(8)));
typedef int   v4i __attribute__((vector_size(16)));   // matches builtin param pointee

#define AS1 __attribute__((address_space(1)))
#define AS3 __attribute__((address_space(3)))

#define NN 2048
#define CC 512
#define FF 512

#define BN 128   // output rows per workgroup
#define BC 64    // output cols per workgroup
#define BK 32    // K chunk
#define LSTR 36  // LDS row stride: rows 16B-aligned (144B) and conflict-free
                 // for 16-row column gathers (gcd(36,64)=4, cycle length 16)

#if __has_builtin(__builtin_amdgcn_global_load_async_to_lds_b128)
#define ASYNC_CP 1
#else
#define ASYNC_CP 0
#endif

__device__ __forceinline__ void cp16(const float* g, float* l) {
#if ASYNC_CP
  // memory -> LDS without VGPR roundtrip; tracked by ASYNCcnt
  __builtin_amdgcn_global_load_async_to_lds_b128(
      (AS1 v4i*)g, (AS3 v4i*)l, 0, 0);
#else
  *(float4*)l = *(const float4*)g;
#endif
}

__device__ __forceinline__ void cp_wait() {
#if ASYNC_CP
#if __has_builtin(__builtin_amdgcn_s_wait_asynccnt)
  __builtin_amdgcn_s_wait_asynccnt(0);
#else
  asm volatile("s_wait_asynccnt 0x0" ::: "memory");
#endif
#endif
}

// -------- norms: one wave (32 threads) per row; xnorm[n] = sum_f x[n][f]^2 ----
__global__ __launch_bounds__(32) void ncm_row_norms(
    const float* __restrict__ x, const float* __restrict__ means,
    float* __restrict__ xnorm, float* __restrict__ mnorm) {
  int row = blockIdx.x;
  const float* src;
  float* dst;
  int r;
  if (row < NN) { src = x + (size_t)row * FF; dst = xnorm; r = row; }
  else          { src = means + (size_t)(row - NN) * FF; dst = mnorm; r = row - NN; }
  float s = 0.f;
  for (int i = threadIdx.x; i < FF; i += 32) {
    float v = src[i];
    s = fmaf(v, v, s);
  }
#pragma unroll
  for (int off = 16; off > 0; off >>= 1) s += __shfl_down(s, off, 32);
  if (threadIdx.x == 0) dst[r] = s;
}

// -------- main kernel: D = X * M^T via v_wmma_f32_16x16x4_f32 ---------------
// 8 waves arranged 4x2; each wave owns a 32x32 output region = 2x2 WMMA tiles.
// Double-buffered LDS; next K-chunk staged via async load-to-LDS during compute.
__global__ __launch_bounds__(256) void ncm_gemm_wmma(
    const float* __restrict__ x, const float* __restrict__ means,
    const float* __restrict__ xnorm, const float* __restrict__ mnorm,
    float* __restrict__ out) {
  __shared__ float Xs[2][BN * LSTR];
  __shared__ float Ms[2][BC * LSTR];

  const int tid  = threadIdx.x;
  const int wave = tid >> 5;
  const int lane = tid & 31;
  const int l15  = lane & 15;
  const int hi   = lane >> 4;   // 0 for lanes 0-15, 1 for lanes 16-31

  const int nbase = blockIdx.x * BN;
  const int cbase = blockIdx.y * BC;

  const int wr = (wave & 3) * 32;   // wave's 32-row slab within block tile
  const int wc = (wave >> 2) * 32;  // wave's 32-col slab within block tile

  v8f acc00 = {}, acc01 = {}, acc10 = {}, acc11 = {};

  // staging: X tile 128x32 = 1024 float4 -> 4/thread; M tile 64x32 = 512 -> 2/thread
  auto stage = [&](int k0, int buf) {
#pragma unroll
    for (int i = 0; i < 4; ++i) {
      const int f  = tid + i * 256;
      const int r  = f >> 3;
      const int c4 = (f & 7) * 4;
      cp16(&x[(size_t)(nbase + r) * FF + k0 + c4], &Xs[buf][r * LSTR + c4]);
    }
#pragma unroll
    for (int i = 0; i < 2; ++i) {
      const int f  = tid + i * 256;
      const int r  = f >> 3;
      const int c4 = (f & 7) * 4;
      cp16(&means[(size_t)(cbase + r) * FF + k0 + c4], &Ms[buf][r * LSTR + c4]);
    }
  };

  // prologue: fill buffer 0
  stage(0, 0);
  cp_wait();
  __syncthreads();

  for (int k0 = 0; k0 < FF; k0 += BK) {
    const int cur = (k0 >> 5) & 1;
    if (k0 + BK < FF) stage(k0 + BK, cur ^ 1);  // overlap with compute below

    const float* xr0 = &Xs[cur][(wr + l15) * LSTR];
    const float* xr1 = &Xs[cur][(wr + 16 + l15) * LSTR];
    const float* mr0 = &Ms[cur][(wc + l15) * LSTR];
    const float* mr1 = &Ms[cur][(wc + 16 + l15) * LSTR];

#pragma unroll
    for (int kk = 0; kk < BK; kk += 4) {
      const int ak = kk + 2 * hi;  // ISA layout: VGPR v holds K = kk + v + 2*hi
      v2f a0; a0.x = xr0[ak]; a0.y = xr0[ak + 1];
      v2f a1; a1.x = xr1[ak]; a1.y = xr1[ak + 1];
      v2f b0; b0.x = mr0[ak]; b0.y = mr0[ak + 1];
      v2f b1; b1.x = mr1[ak]; b1.y = mr1[ak + 1];
      acc00 = __builtin_amdgcn_wmma_f32_16x16x4_f32(false, a0, false, b0,
                                                    (short)0, acc00, false, false);
      acc01 = __builtin_amdgcn_wmma_f32_16x16x4_f32(false, a0, false, b1,
                                                    (short)0, acc01, false, false);
      acc10 = __builtin_amdgcn_wmma_f32_16x16x4_f32(false, a1, false, b0,
                                                    (short)0, acc10, false, false);
      acc11 = __builtin_amdgcn_wmma_f32_16x16x4_f32(false, a1, false, b1,
                                                    (short)0, acc11, false, false);
    }

    cp_wait();        // own async copies for next chunk have landed in LDS
    __syncthreads();  // everyone's copies visible; current buffer reads done
  }

  // epilogue: out[n][c] = 2*dot - xnorm[n] - mnorm[c]
  const float mn0 = mnorm[cbase + wc + l15];
  const float mn1 = mnorm[cbase + wc + 16 + l15];
#pragma unroll
  for (int v = 0; v < 8; ++v) {
    const int r0 = nbase + wr + v + 8 * hi;        // C/D layout: VGPR v, half 'hi'
    const int r1 = r0 + 16;
    const float xn0 = xnorm[r0];
    const float xn1 = xnorm[r1];
    float* o0 = out + (size_t)r0 * CC + cbase + wc;
    float* o1 = out + (size_t)r1 * CC + cbase + wc;
    o0[l15]      = 2.f * acc00[v] - xn0 - mn0;
    o0[16 + l15] = 2.f * acc01[v] - xn0 - mn1;
    o1[l15]      = 2.f * acc10[v] - xn1 - mn0;
    o1[16 + l15] = 2.f * acc11[v] - xn1 - mn1;
  }
}

extern "C" void kernel_launch(void* const* d_in, const int* in_sizes, int n_in,
                              void* d_out, int out_size, void* d_ws, size_t ws_size,
                              hipStream_t stream) {
  const float* x     = (const float*)d_in[0];
  const float* means = (const float*)d_in[1];
  float* out   = (float*)d_out;
  float* xnorm = (float*)d_ws;          // NN floats
  float* mnorm = xnorm + NN;            // CC floats

  ncm_row_norms<<<NN + CC, 32, 0, stream>>>(x, means, xnorm, mnorm);

  dim3 grid(NN / BN, CC / BC);  // 16 x 8 workgroups, 8 waves each
  ncm_gemm_wmma<<<grid, 256, 0, stream>>>(x, means, xnorm, mnorm, out);
}